// Actor_55997783605447
// MI455X (gfx1250) — compile-verified
//
#include <hip/hip_runtime.h>

constexpr int kNodes  = 50000;
constexpr int kEdges  = 800000;
constexpr int kFea    = 128;
constexpr int kLayers = 3;
constexpr int kHid    = 512;
constexpr int kGraphs = 64;

typedef __attribute__((ext_vector_type(2))) float v2f;
typedef __attribute__((ext_vector_type(8))) float v8f;

static __device__ __forceinline__ void atomic_add_f32(float* p, float v) {
    __hip_atomic_fetch_add(p, v, __ATOMIC_RELAXED, __HIP_MEMORY_SCOPE_AGENT);
}

// ---------------------------------------------------------------------------
// WMMA GEMM: C[M,N] = A[M,K] @ B[K,N] (+bias) (+relu), all row-major.
// K % 16 == 0, N % 32 == 0. One wave computes a 16x32 tile (two 16x16
// accumulators sharing one A fragment -> two independent wmma chains).
// 4 waves per block cover 64 rows.
// Fragment layouts per CDNA5 ISA 7.12.2:
//   A 16x4 f32: lanes 0-15 rows M; vgpr{0,1} = K{+0,+1} (lo half) / K{+2,+3} (hi)
//   D 16x16:    lane -> column n0+(lane&15), vgpr v -> row m0 + 8*(lane>>4) + v
// Out-of-range rows: clamp pointer to a valid row and multiply fragment by 0
// (no EXEC divergence; WMMA requires EXEC all-ones).
// ---------------------------------------------------------------------------
__global__ void gemm_wmma(const float* __restrict__ A, int lda,
                          const float* __restrict__ B, int ldb,
                          float* __restrict__ C, int ldc,
                          const float* __restrict__ bias,
                          int M, int K, int do_relu)
{
    const int lane = threadIdx.x & 31;
    const int wave = threadIdx.x >> 5;
    const int m0   = (blockIdx.x * 4 + wave) * 16;
    const int n0   = blockIdx.y * 32;          // 32 columns per wave
    const int half = lane >> 4;                // 0: lanes 0-15, 1: lanes 16-31
    const int l15  = lane & 15;

    const int   rowA = m0 + l15;
    const float mask = (rowA < M) ? 1.0f : 0.0f;   // zero out-of-range rows
    const int   rowS = (rowA < M) ? rowA : (M - 1);
    const float* __restrict__ arow = A + (long)rowS * lda + half * 2;
    const float* __restrict__ bp0  = B + (long)(half * 2) * ldb + n0 + l15;
    const float* __restrict__ bp1  = bp0 + 16;

    v8f acc0 = {}, acc1 = {};
    for (int k = 0; k < K; k += 16) {
        #pragma unroll
        for (int u = 0; u < 4; ++u) {
            const int  kk = k + 4 * u;
            v2f a = *(const v2f*)(arow + kk);   // 8B-aligned b64 load
            a.x *= mask;
            a.y *= mask;
            const long o0 = (long)kk * ldb;
            v2f b0, b1;
            b0.x = bp0[o0];
            b0.y = bp0[o0 + ldb];
            b1.x = bp1[o0];
            b1.y = bp1[o0 + ldb];
            acc0 = __builtin_amdgcn_wmma_f32_16x16x4_f32(
                false, a, false, b0, (short)0, acc0, false, false);
            acc1 = __builtin_amdgcn_wmma_f32_16x16x4_f32(
                false, a, false, b1, (short)0, acc1, false, false);
        }
    }

    const int   col0 = n0 + l15, col1 = col0 + 16;
    const float bv0  = bias ? bias[col0] : 0.0f;
    const float bv1  = bias ? bias[col1] : 0.0f;
    #pragma unroll
    for (int v = 0; v < 8; ++v) {
        const int row = m0 + half * 8 + v;
        if (row < M) {
            float c0 = acc0[v] + bv0;
            float c1 = acc1[v] + bv1;
            if (do_relu) { c0 = fmaxf(c0, 0.0f); c1 = fmaxf(c1, 0.0f); }
            C[(long)row * ldc + col0] = c0;
            C[(long)row * ldc + col1] = c1;
        }
    }
}

// --------------------------- degree / dinv ---------------------------------
__global__ void k_deg_init(float* __restrict__ deg) {
    int i = blockIdx.x * blockDim.x + threadIdx.x;
    if (i < kNodes) deg[i] = 1.0f;   // self-loop
}
__global__ void k_deg_count(const int* __restrict__ dst, float* __restrict__ deg) {
    int e = blockIdx.x * blockDim.x + threadIdx.x;
    if (e < kEdges) atomic_add_f32(&deg[dst[e]], 1.0f);
}
__global__ void k_deg_rsqrt(float* __restrict__ deg) {
    int i = blockIdx.x * blockDim.x + threadIdx.x;
    if (i < kNodes) deg[i] = rsqrtf(deg[i]);   // in-place -> dinv
}

// ------------------- self-loop term + bias init of agg ---------------------
__global__ void k_self_init(const float4* __restrict__ h4, const float* __restrict__ dinv,
                            const float4* __restrict__ bias4, float4* __restrict__ agg4) {
    int i = blockIdx.x * blockDim.x + threadIdx.x;   // over NF/4
    if (i < kNodes * kFea / 4) {
        const float d = dinv[i >> 5];
        const float s = d * d;
        const float4 hv = h4[i];
        const float4 bv = bias4[i & 31];
        agg4[i] = make_float4(hv.x * s + bv.x, hv.y * s + bv.y,
                              hv.z * s + bv.z, hv.w * s + bv.w);
    }
}

// ---------------- edge gather-scale-scatter (one wave per edge) ------------
// h rows are 512B; a wave loads one row as 32 x b128 (fully coalesced) and
// scatters 128 fp32 HW atomics into agg[dst]. Both h and agg are L2-resident
// (25.6 MB each in 192 MB L2), so this runs at L2 rates, not HBM.
__global__ void k_edge_scatter(const int* __restrict__ src, const int* __restrict__ dst,
                               const float* __restrict__ dinv, const float* __restrict__ h,
                               float* __restrict__ agg) {
    int t = blockIdx.x * blockDim.x + threadIdx.x;
    int e = t >> 5;
    if (e >= kEdges) return;
    const int lane = t & 31;
    const int s = src[e], d = dst[e];
    const float coef = dinv[s] * dinv[d];
    const float4 hv = ((const float4*)(h + (long)s * kFea))[lane];
    float* __restrict__ ap = agg + (long)d * kFea + lane * 4;
    atomic_add_f32(ap + 0, hv.x * coef);
    atomic_add_f32(ap + 1, hv.y * coef);
    atomic_add_f32(ap + 2, hv.z * coef);
    atomic_add_f32(ap + 3, hv.w * coef);
}

__global__ void k_relu(const float4* __restrict__ in, float4* __restrict__ out) {
    int i = blockIdx.x * blockDim.x + threadIdx.x;
    if (i < kNodes * kFea / 4) {
        const float4 v = in[i];
        out[i] = make_float4(fmaxf(v.x, 0.0f), fmaxf(v.y, 0.0f),
                             fmaxf(v.z, 0.0f), fmaxf(v.w, 0.0f));
    }
}

// ----------------------- node head: x @ node_w + b -------------------------
__global__ void k_node_head(const float* __restrict__ x, const float* __restrict__ w,
                            const float* __restrict__ b, float* __restrict__ out) {
    int t = blockIdx.x * blockDim.x + threadIdx.x;
    int node = t >> 5;
    if (node >= kNodes) return;
    const int lane = t & 31;
    const float4 xv = ((const float4*)(x + (long)node * kFea))[lane];
    const float4 wv = ((const float4*)w)[lane];
    float acc = xv.x * wv.x + xv.y * wv.y + xv.z * wv.z + xv.w * wv.w;
    #pragma unroll
    for (int off = 16; off > 0; off >>= 1)
        acc += __shfl_down(acc, off, 32);
    if (lane == 0) out[node] = acc + b[0];
}

// ------------------------------ mean pool ----------------------------------
__global__ void k_pool_scatter(const float* __restrict__ x, const int* __restrict__ batch,
                               float* __restrict__ sums, float* __restrict__ cnts) {
    int t = blockIdx.x * blockDim.x + threadIdx.x;
    int node = t >> 5;
    if (node >= kNodes) return;
    const int lane = t & 31;
    const int g = batch[node];
    const float4 xv = ((const float4*)(x + (long)node * kFea))[lane];
    float* __restrict__ sr = sums + (long)g * kFea + lane * 4;
    atomic_add_f32(sr + 0, xv.x);
    atomic_add_f32(sr + 1, xv.y);
    atomic_add_f32(sr + 2, xv.z);
    atomic_add_f32(sr + 3, xv.w);
    if (lane == 0) atomic_add_f32(&cnts[g], 1.0f);
}
__global__ void k_pool_div(const float* __restrict__ sums, const float* __restrict__ cnts,
                           float* __restrict__ fea) {
    int i = blockIdx.x * blockDim.x + threadIdx.x;
    if (i < kGraphs * kFea) fea[i] = sums[i] / fmaxf(cnts[i >> 7], 1.0f);
}

// ---------------------------------------------------------------------------
extern "C" void kernel_launch(void* const* d_in, const int* in_sizes, int n_in,
                              void* d_out, int out_size, void* d_ws, size_t ws_size,
                              hipStream_t stream) {
    const float* x_in   = (const float*)d_in[0];
    const int*   eidx   = (const int*)d_in[1];
    const int*   batch  = (const int*)d_in[2];
    const float* conv_W = (const float*)d_in[3];
    const float* conv_b = (const float*)d_in[4];
    const float* fw1    = (const float*)d_in[5];
    const float* fb1    = (const float*)d_in[6];
    const float* fw2    = (const float*)d_in[7];
    const float* fb2    = (const float*)d_in[8];
    const float* nw     = (const float*)d_in[9];
    const float* nb     = (const float*)d_in[10];
    float* out = (float*)d_out;

    const int* src = eidx;            // edge_index[0]
    const int* dst = eidx + kEdges;   // edge_index[1]

    // workspace layout (floats)
    float* ws   = (float*)d_ws;
    const int NF = kNodes * kFea;               // 6.4M
    float* xbuf = ws;                           // NF
    float* hbuf = xbuf + NF;                    // NF
    float* abuf = hbuf + NF;                    // NF
    float* dinv = abuf + NF;                    // 50048 (padded)
    float* sums = dinv + 50048;                 // kGraphs*kFea
    float* cnts = sums + kGraphs * kFea;        // 64 (contiguous after sums)
    float* fea  = cnts + 64;                    // kGraphs*kFea
    float* fea1 = fea  + kGraphs * kFea;        // kGraphs*kHid

    // --- degrees (layer-independent) ---
    k_deg_init <<<(kNodes + 255) / 256, 256, 0, stream>>>(dinv);
    k_deg_count<<<(kEdges + 255) / 256, 256, 0, stream>>>(dst, dinv);
    k_deg_rsqrt<<<(kNodes + 255) / 256, 256, 0, stream>>>(dinv);

    const dim3 gblk(128);   // 4 waves of 32
    for (int l = 0; l < kLayers; ++l) {
        const float* xin = (l == 0) ? x_in : xbuf;
        dim3 ggrid((kNodes + 63) / 64, kFea / 32);
        gemm_wmma<<<ggrid, gblk, 0, stream>>>(xin, kFea,
                                              conv_W + (long)l * kFea * kFea, kFea,
                                              hbuf, kFea, nullptr, kNodes, kFea, 0);
        k_self_init   <<<(NF / 4 + 255) / 256, 256, 0, stream>>>(
            (const float4*)hbuf, dinv, (const float4*)(conv_b + (long)l * kFea),
            (float4*)abuf);
        k_edge_scatter<<<(kEdges * 32 + 255) / 256, 256, 0, stream>>>(src, dst, dinv,
                                                                      hbuf, abuf);
        k_relu        <<<(NF / 4 + 255) / 256, 256, 0, stream>>>((const float4*)abuf,
                                                                 (float4*)xbuf);
    }

    // node_prob -> out[0 .. kNodes)
    k_node_head<<<(kNodes * 32 + 255) / 256, 256, 0, stream>>>(xbuf, nw, nb, out);

    // global mean pool
    hipMemsetAsync(sums, 0, (size_t)(kGraphs * kFea + 64) * sizeof(float), stream);
    k_pool_scatter<<<(kNodes * 32 + 255) / 256, 256, 0, stream>>>(xbuf, batch, sums, cnts);
    k_pool_div<<<(kGraphs * kFea + 255) / 256, 256, 0, stream>>>(sums, cnts, fea);

    // fea MLP heads (WMMA)
    dim3 g1((kGraphs + 63) / 64, kHid / 32);
    gemm_wmma<<<g1, gblk, 0, stream>>>(fea, kFea, fw1, kHid, fea1, kHid, fb1,
                                       kGraphs, kFea, 1);
    dim3 g2((kGraphs + 63) / 64, kFea / 32);
    gemm_wmma<<<g2, gblk, 0, stream>>>(fea1, kHid, fw2, kFea, out + kNodes, kFea, fb2,
                                       kGraphs, kHid, 0);
}